// FarthestPointSample_11879879542890
// MI455X (gfx1250) — compile-verified
//
#include <hip/hip_runtime.h>
#include <stdint.h>

// ---------------------------------------------------------------------------
// Farthest Point Sampling, MI455X (gfx1250).
// 1023 sequentially dependent global argmax rounds => latency/sync bound.
// Points + running distances live in VGPRs; per-round pass is pure VALU.
// CDNA5 paths: async global->LDS staging (ASYNCcnt), monitor-load + 
// s_monitor_sleep spin barrier, wave32 shfl reductions, u64 atomic argmax.
// WMMA is deliberately absent: per round this is a [N,3]x[3,1] matvec with a
// serial dependency; wave32 VALU yields 32 dists / ~8 ops vs 16 useful dots
// per V_WMMA_F32_16X16X4_F32 plus permute cleanup -- WMMA would be slower.
// ---------------------------------------------------------------------------

#define BATCHES 8
#define NPTS    131072
#define NWGPB   8                    // workgroups cooperating per batch
#define TPB     1024                 // threads per workgroup (32 waves)
#define PPT     16                   // points per thread (in registers)
#define LPW     (TPB * PPT)          // 16384 points per workgroup
#define CHUNKP  4096                 // staging chunk: 4096 pts * 12B = 48KB
#define NCHUNK  (LPW / CHUNKP)       // 4

// ws layout, cacheline-isolated (256B units):
//   block A, byte b*256   : cnt (u32 @+0), slot (u64 @+8), bc[3] (u32 @+16..27)
//   block B, byte 4096+b*256 : flag (u32) -- written ONLY by release store,
//                              so the monitored line stays quiet until release.
#define WS_U32_TOTAL 1536            // 6 KB control region

__device__ __forceinline__ void async_copy_b128(unsigned ldsByteAddr, const void* gaddr) {
    asm volatile("global_load_async_to_lds_b128 %0, %1, off"
                 :: "v"(ldsByteAddr), "v"((unsigned long long)(uintptr_t)gaddr)
                 : "memory");
}

__device__ __forceinline__ void wait_async_zero() {
#if __has_builtin(__builtin_amdgcn_s_wait_asynccnt)
    __builtin_amdgcn_s_wait_asynccnt(0);
#else
    asm volatile("s_wait_asynccnt 0" ::: "memory");
#endif
}

// CDNA5 MWAIT: device-scope monitor load arms GL2 to wake this wave when the
// cacheline is written; s_monitor_sleep parks until then (or timeout).
__device__ __forceinline__ unsigned monitor_load_dev(const unsigned* p) {
    unsigned v;
    asm volatile("global_load_monitor_b32 %0, %1, off scope:SCOPE_DEV\n\t"
                 "s_wait_loadcnt 0x0"
                 : "=v"(v)
                 : "v"((unsigned long long)(uintptr_t)p)
                 : "memory");
    return v;
}

__device__ __forceinline__ void monitor_sleep() {
    asm volatile("s_monitor_sleep 8" ::: "memory");
}

__global__ void fps_init(unsigned* __restrict__ ws) {
    for (int i = (int)threadIdx.x; i < WS_U32_TOTAL; i += (int)blockDim.x)
        ws[i] = 0u;
}

__global__ __launch_bounds__(TPB, 1)
void fps_main(const float* __restrict__ xyz,
              const int*   __restrict__ npoint_p,
              int*         __restrict__ out,
              unsigned*    __restrict__ ws)
{
    __shared__ float               s_stage[CHUNKP * 3];   // 48 KB staging
    __shared__ unsigned long long  s_red[TPB / 32];
    __shared__ float               s_c[3];

    const int t   = (int)threadIdx.x;
    const int bat = (int)blockIdx.x / NWGPB;
    const int w   = (int)blockIdx.x % NWGPB;
    const int np  = npoint_p[0];                          // 1024

    unsigned*            cnt  = ws + bat * 64;                           // byte b*256
    unsigned long long*  slot = (unsigned long long*)(ws + bat * 64 + 2);
    unsigned*            bc   = ws + bat * 64 + 4;
    unsigned*            flag = ws + 1024 + bat * 64;                    // byte 4096+b*256

    const float* bx = xyz + (size_t)bat * NPTS * 3;

    float px[PPT], py[PPT], pz[PPT], pd[PPT];

    // ---- One-time staging: coalesced async global->LDS b128, then to VGPRs.
    const unsigned ldsBase = (unsigned)(uintptr_t)&s_stage[0]; // low 32 bits = LDS offset
    for (int c = 0; c < NCHUNK; ++c) {
        const char* gbase = (const char*)(bx + ((size_t)w * LPW + (size_t)c * CHUNKP) * 3);
        #pragma unroll
        for (int j = 0; j < 3; ++j) {                      // 3 * 1024 * 16B = 48KB
            unsigned off = (unsigned)((j * TPB + t) * 16);
            async_copy_b128(ldsBase + off, gbase + off);
        }
        wait_async_zero();                                 // my wave's copies done
        __syncthreads();                                   // everyone's copies done
        #pragma unroll
        for (int kk = 0; kk < PPT / NCHUNK; ++kk) {
            int k = c * (PPT / NCHUNK) + kk;
            int q = kk * TPB + t;                          // 12B stride: LDS conflict-free
            px[k] = s_stage[q * 3 + 0];
            py[k] = s_stage[q * 3 + 1];
            pz[k] = s_stage[q * 3 + 2];
            pd[k] = 1e10f;
        }
        __syncthreads();
    }

    // ---- Round 0: centroid = point 0; reference emits index 0 first.
    if (t == 0) {
        s_c[0] = bx[0]; s_c[1] = bx[1]; s_c[2] = bx[2];
        if (w == 0) out[(size_t)bat * np + 0] = 0;
    }
    __syncthreads();
    float cx = s_c[0], cy = s_c[1], cz = s_c[2];

    const unsigned baseIdx = (unsigned)(w * LPW + t);

    for (int it = 1; it < np; ++it) {
        // -- register-resident distance update + thread-local argmax
        float bd = -1.0f; unsigned bi = 0u;
        #pragma unroll
        for (int k = 0; k < PPT; ++k) {
            // no FMA contraction: match reference mul-then-add rounding exactly
            float dx = __fsub_rn(px[k], cx);
            float dy = __fsub_rn(py[k], cy);
            float dz = __fsub_rn(pz[k], cz);
            float d  = __fadd_rn(__fadd_rn(__fmul_rn(dx, dx), __fmul_rn(dy, dy)),
                                 __fmul_rn(dz, dz));
            float nd = fminf(pd[k], d);
            pd[k] = nd;
            if (nd > bd) { bd = nd; bi = baseIdx + (unsigned)(k * TPB); }
        }
        // packed key: bigger dist wins; equal dist -> smaller index wins
        unsigned long long key =
            ((unsigned long long)__float_as_uint(bd) << 32) |
            (unsigned long long)(0xFFFFFFFFu - bi);

        // -- wave32 shuffle reduction
        #pragma unroll
        for (int m = 16; m >= 1; m >>= 1) {
            unsigned long long o = __shfl_xor(key, m, 32);
            key = (o > key) ? o : key;
        }
        if ((t & 31) == 0) s_red[t >> 5] = key;
        __syncthreads();

        if (t < TPB / 32) {                                // wave 0 reduces 32 waves
            key = s_red[t];
            #pragma unroll
            for (int m = 16; m >= 1; m >>= 1) {
                unsigned long long o = __shfl_xor(key, m, 32);
                key = (o > key) ? o : key;
            }
            if (t == 0) {
                // -- cross-WGP combine: one u64 atomic max per workgroup
                __hip_atomic_fetch_max(slot, key,
                                       __ATOMIC_ACQ_REL, __HIP_MEMORY_SCOPE_AGENT);
                unsigned prev = __hip_atomic_fetch_add(cnt, 1u,
                                       __ATOMIC_ACQ_REL, __HIP_MEMORY_SCOPE_AGENT);
                if (prev == (unsigned)(NWGPB - 1)) {
                    // last arriver: decode winner, publish, reset, release
                    unsigned long long wk = __hip_atomic_load(slot,
                                       __ATOMIC_RELAXED, __HIP_MEMORY_SCOPE_AGENT);
                    unsigned widx = 0xFFFFFFFFu - (unsigned)(wk & 0xFFFFFFFFull);
                    out[(size_t)bat * np + it] = (int)widx;
                    const float* cp = bx + (size_t)widx * 3;
                    __hip_atomic_store(&bc[0], __float_as_uint(cp[0]),
                                       __ATOMIC_RELAXED, __HIP_MEMORY_SCOPE_AGENT);
                    __hip_atomic_store(&bc[1], __float_as_uint(cp[1]),
                                       __ATOMIC_RELAXED, __HIP_MEMORY_SCOPE_AGENT);
                    __hip_atomic_store(&bc[2], __float_as_uint(cp[2]),
                                       __ATOMIC_RELAXED, __HIP_MEMORY_SCOPE_AGENT);
                    __hip_atomic_store(slot, 0ull,
                                       __ATOMIC_RELAXED, __HIP_MEMORY_SCOPE_AGENT);
                    __hip_atomic_store(cnt, 0u,
                                       __ATOMIC_RELAXED, __HIP_MEMORY_SCOPE_AGENT);
                    __hip_atomic_store(flag, (unsigned)it,
                                       __ATOMIC_RELEASE, __HIP_MEMORY_SCOPE_AGENT);
                } else {
                    // CDNA5 MWAIT spin: park until GL2 reports the flag line
                    // was written (release store above), then acquire-fence.
                    while (monitor_load_dev(flag) < (unsigned)it)
                        monitor_sleep();
                    __builtin_amdgcn_fence(__ATOMIC_ACQUIRE, "agent");
                }
                s_c[0] = __uint_as_float(__hip_atomic_load(&bc[0],
                             __ATOMIC_RELAXED, __HIP_MEMORY_SCOPE_AGENT));
                s_c[1] = __uint_as_float(__hip_atomic_load(&bc[1],
                             __ATOMIC_RELAXED, __HIP_MEMORY_SCOPE_AGENT));
                s_c[2] = __uint_as_float(__hip_atomic_load(&bc[2],
                             __ATOMIC_RELAXED, __HIP_MEMORY_SCOPE_AGENT));
            }
        }
        __syncthreads();
        cx = s_c[0]; cy = s_c[1]; cz = s_c[2];
    }
}

extern "C" void kernel_launch(void* const* d_in, const int* in_sizes, int n_in,
                              void* d_out, int out_size, void* d_ws, size_t ws_size,
                              hipStream_t stream) {
    (void)in_sizes; (void)n_in; (void)out_size; (void)ws_size;
    const float* xyz    = (const float*)d_in[0];
    const int*   npoint = (const int*)d_in[1];
    int*         out    = (int*)d_out;           // reference output dtype is int32
    unsigned*    ws     = (unsigned*)d_ws;

    hipLaunchKernelGGL(fps_init, dim3(1), dim3(1024), 0, stream, ws);
    hipLaunchKernelGGL(fps_main, dim3(BATCHES * NWGPB), dim3(TPB), 0, stream,
                       xyz, npoint, out, ws);
}